// PointNetSetAbstractionMsg_72567767433504
// MI455X (gfx1250) — compile-verified
//
#include <hip/hip_runtime.h>
#include <math.h>

// ---------------------------------------------------------------------------
// PointNet++ Set Abstraction (MSG) for MI455X / gfx1250.
//  - FPS: serial 512-step argmax per batch (one 1024-thread block per batch,
//    all state in LDS).
//  - Per radius: fused ball-query -> 3x (WMMA f32 16x16x4 GEMM + folded
//    BN/bias/ReLU) -> max-over-K, entirely in LDS. One workgroup owns 128
//    GEMM columns. Matrix ops: V_WMMA_F32_16X16X4_F32 (fp32 = ref dtype).
// ---------------------------------------------------------------------------

typedef __attribute__((ext_vector_type(2))) float v2f;
typedef __attribute__((ext_vector_type(4))) float v4f;
typedef __attribute__((ext_vector_type(8))) float v8f;

#define NPTS   4096
#define S_CENT 512
#define BATCH  8
#define LD     136          // LDS row stride in floats (kills bank overlap)
#define BN_EPS 1e-5f

struct LayerP { const float *w, *b, *g, *be, *mu, *va; };
struct BranchP { LayerP l1, l2, l3; };

// ---------------------------------------------------------------------------
// Farthest point sampling: one block per batch. Matches jax scan semantics:
// record current farthest, update min-distance, argmax (first-index tie-break).
// ---------------------------------------------------------------------------
__global__ void __launch_bounds__(1024) fps_kernel(const float* __restrict__ xyz,
                                                   float* __restrict__ nxyz,
                                                   float* __restrict__ out) {
    __shared__ float sx[NPTS], sy[NPTS], sz[NPTS], dist[NPTS];
    __shared__ float rval[1024];
    __shared__ int   ridx[1024];

    const int b = blockIdx.x, tid = threadIdx.x;
    const float* X = xyz + (size_t)b * 3 * NPTS;

    for (int p = tid; p < NPTS; p += 1024) {
        sx[p] = X[p]; sy[p] = X[NPTS + p]; sz[p] = X[2 * NPTS + p];
        dist[p] = 1e10f;
    }
    __syncthreads();

    int far = 0;
    for (int s = 0; s < S_CENT; ++s) {
        const float cx = sx[far], cy = sy[far], cz = sz[far];
        if (tid == 0) {
            float* nz = nxyz + ((size_t)b * S_CENT + s) * 3;
            nz[0] = cx; nz[1] = cy; nz[2] = cz;
            float* o = out + (size_t)b * 3 * S_CENT;   // new_xyz, (B,3,S)
            o[s] = cx; o[S_CENT + s] = cy; o[2 * S_CENT + s] = cz;
        }
        float bv = -1.f; int bi = 0;
        #pragma unroll
        for (int j = 0; j < NPTS / 1024; ++j) {
            const int p = tid + j * 1024;
            const float dx = sx[p] - cx, dy = sy[p] - cy, dz = sz[p] - cz;
            const float d  = dx * dx + dy * dy + dz * dz;
            const float nd = fminf(dist[p], d);
            dist[p] = nd;
            if (nd > bv) { bv = nd; bi = p; }   // strictly-greater keeps lowest idx
        }
        rval[tid] = bv; ridx[tid] = bi;
        __syncthreads();
        for (int off = 512; off > 0; off >>= 1) {
            if (tid < off) {
                const float v2 = rval[tid + off]; const int i2 = ridx[tid + off];
                if (v2 > rval[tid] || (v2 == rval[tid] && i2 < ridx[tid])) {
                    rval[tid] = v2; ridx[tid] = i2;
                }
            }
            __syncthreads();
        }
        far = ridx[0];
        __syncthreads();   // protect rval/ridx reuse next iteration
    }
}

// ---------------------------------------------------------------------------
// One MLP layer on a 128-column LDS tile via V_WMMA_F32_16X16X4_F32.
// in: icp x LD (padded rows zeroed), out: oc x LD. 8 waves split output tiles.
// scl/shf: per-channel folded BN constants in LDS.
// ISA f32 layouts: A(16x4): lane L holds M=L%16, K = {0,1}(+2 if L>=16).
// C/D(16x16): VGPR i, lane L -> M = i + (L<16 ? 0 : 8), N = L%16.
// Call sites pass compile-time ic/icp/oc so the pad select folds away when
// ic == icp (layers 2 and 3).
// ---------------------------------------------------------------------------
__device__ __forceinline__ void mlp_layer(const float* __restrict__ inL,
                                          float* __restrict__ outL,
                                          const float* __restrict__ W,
                                          const float* __restrict__ scl,
                                          const float* __restrict__ shf,
                                          int ic, int icp, int oc,
                                          int wv, int lane) {
    const int Mtiles = oc >> 4;
    const int total  = Mtiles * 8;         // 8 column tiles of 16
    const int m  = lane & 15;
    const int hi = lane >> 4;
    for (int t = wv; t < total; t += 8) {
        const int mt  = t >> 3;
        const int nt  = t & 7;
        const int col = nt * 16 + m;
        const int row = mt * 16 + m;
        v8f acc = {0.f, 0.f, 0.f, 0.f, 0.f, 0.f, 0.f, 0.f};
        for (int k0 = 0; k0 < icp; k0 += 4) {
            const int kk = k0 + (hi << 1);
            // Clamped unconditional b64 load + select (no exec branching);
            // select folds away when ic == icp.
            const int kc = (kk < ic) ? kk : 0;
            v2f a = *(const v2f*)(W + row * ic + kc);
            if (kk >= ic) { a.x = 0.f; a.y = 0.f; }
            v2f bb;
            bb.x = inL[kk * LD + col];
            bb.y = inL[(kk + 1) * LD + col];
            acc = __builtin_amdgcn_wmma_f32_16x16x4_f32(
                false, a, false, bb, (short)0, acc, false, false);
        }
        // Fused (bias+BN folded) scale/shift + ReLU epilogue, vectorized.
        const int rb = mt * 16 + hi * 8;   // 8-aligned -> 16B-aligned LDS reads
        const v4f s0 = *(const v4f*)(scl + rb);
        const v4f s1 = *(const v4f*)(scl + rb + 4);
        const v4f f0 = *(const v4f*)(shf + rb);
        const v4f f1 = *(const v4f*)(shf + rb + 4);
        #pragma unroll
        for (int i = 0; i < 4; ++i)
            outL[(rb + i) * LD + col]     = fmaxf(acc[i]     * s0[i] + f0[i], 0.f);
        #pragma unroll
        for (int i = 0; i < 4; ++i)
            outL[(rb + 4 + i) * LD + col] = fmaxf(acc[4 + i] * s1[i] + f1[i], 0.f);
    }
}

// ---------------------------------------------------------------------------
// Fused branch kernel: ball query -> grouped features -> 3 WMMA layers -> max.
// One workgroup (256 threads, 8 waves) owns CPW = 128/K centers (128 columns).
// ---------------------------------------------------------------------------
template<int K, int C1, int C2, int C3>
__global__ void __launch_bounds__(256) msg_branch(const float* __restrict__ xyz,
                                                  const float* __restrict__ pts,
                                                  const float* __restrict__ nxyz,
                                                  float* __restrict__ out,
                                                  BranchP P, float r2, int chBase) {
    constexpr int CPW = 128 / K;
    constexpr int CT  = C1 + C2 + C3;
    __shared__ float bufA[C2 * LD];     // layer1 input (8 rows) / layer2 output
    __shared__ float bufB[C3 * LD];     // layer1 output / layer3 output
    __shared__ float scl[CT], shf[CT];  // folded bias+BN per output channel
    __shared__ int   idxb[128];
    __shared__ float cctr[CPW][3];

    const int tid  = threadIdx.x;
    const int lane = tid & 31, wv = tid >> 5;
    const int gc0  = blockIdx.x * CPW;              // first flattened (b*S+s)
    const int b    = gc0 / S_CENT;                  // CPW divides S -> same batch
    const float* X  = xyz + (size_t)b * 3 * NPTS;
    const float* Pp = pts + (size_t)b * 3 * NPTS;

    // ---- Phase 1: ball query, one wave per center (ordered first-K collect).
    if (wv < CPW) {
        const int gc = gc0 + wv;
        const float* nz = nxyz + (size_t)gc * 3;
        const float cx = nz[0], cy = nz[1], cz = nz[2];
        if (lane == 0) { cctr[wv][0] = cx; cctr[wv][1] = cy; cctr[wv][2] = cz; }
        const float cn = cx * cx + cy * cy + cz * cz;
        int cnt = 0, firstIdx = 0;
        bool haveFirst = false;
        for (int n0 = 0; n0 < NPTS; n0 += 32) {
            if (cnt >= K) break;
            const int n = n0 + lane;
            const float x = X[n], y = X[NPTS + n], z = X[2 * NPTS + n];
            const float pn = x * x + y * y + z * z;
            const float d  = cn + pn - 2.f * (cx * x + cy * y + cz * z);
            const bool pred = d <= r2;               // sqrd > r^2 -> excluded
            const unsigned mm = (unsigned)__ballot(pred);
            if (!haveFirst && mm) { firstIdx = n0 + __ffs(mm) - 1; haveFirst = true; }
            const int pos = cnt + __popc(mm & ((1u << lane) - 1u));
            if (pred && pos < K) idxb[wv * K + pos] = n;
            cnt += __popc(mm);
        }
        if (cnt > K) cnt = K;
        for (int p2 = cnt + lane; p2 < K; p2 += 32) idxb[wv * K + p2] = firstIdx;
    }

    // ---- Phase 1b (all threads, overlaps with ball query waves finishing):
    // fold bias + batchnorm into per-channel scale/shift:
    //   h = acc*sc + ((b - mu)*sc + beta),  sc = gamma / sqrt(var + eps)
    for (int e = tid; e < CT; e += 256) {
        const LayerP* L; int r;
        if (e < C1)           { L = &P.l1; r = e; }
        else if (e < C1 + C2) { L = &P.l2; r = e - C1; }
        else                  { L = &P.l3; r = e - C1 - C2; }
        const float sc = L->g[r] / sqrtf(L->va[r] + BN_EPS);
        scl[e] = sc;
        shf[e] = (L->b[r] - L->mu[r]) * sc + L->be[r];
    }
    __syncthreads();

    // ---- Phase 2: gather [points(3), xyz - center(3)] into bufA rows 0..7.
    for (int e = tid; e < 8 * 128; e += 256) {
        const int ch = e >> 7, col = e & 127;
        float val = 0.f;                             // rows 6,7 are zero padding
        if (ch < 6) {
            const int idx = idxb[col];
            if (ch < 3) val = Pp[ch * NPTS + idx];
            else {
                const int cw = col / K;
                val = X[(ch - 3) * NPTS + idx] - cctr[cw][ch - 3];
            }
        }
        bufA[ch * LD + col] = val;
    }
    __syncthreads();

    // ---- Phase 3: three fused WMMA GEMM + scale/shift + ReLU layers.
    mlp_layer(bufA, bufB, P.l1.w, scl,           shf,           6,  8,  C1, wv, lane);
    __syncthreads();
    mlp_layer(bufB, bufA, P.l2.w, scl + C1,      shf + C1,      C1, C1, C2, wv, lane);
    __syncthreads();
    mlp_layer(bufA, bufB, P.l3.w, scl + C1 + C2, shf + C1 + C2, C2, C2, C3, wv, lane);
    __syncthreads();

    // ---- Phase 4: max over K samples, write (B, 320, S) feature block.
    for (int e = tid; e < C3 * CPW; e += 256) {
        const int ch = e / CPW, cw = e % CPW;
        const float* row = bufB + ch * LD + cw * K;
        float mx = row[0];
        #pragma unroll 4
        for (int k = 1; k < K; ++k) mx = fmaxf(mx, row[k]);
        const int gc = gc0 + cw;
        const int bb = gc / S_CENT, s = gc % S_CENT;
        out[(size_t)BATCH * 3 * S_CENT +
            ((size_t)bb * 320 + chBase + ch) * S_CENT + s] = mx;
    }
}

// ---------------------------------------------------------------------------
extern "C" void kernel_launch(void* const* d_in, const int* in_sizes, int n_in,
                              void* d_out, int out_size, void* d_ws, size_t ws_size,
                              hipStream_t stream) {
    (void)in_sizes; (void)n_in; (void)out_size; (void)ws_size;
    const float* xyz = (const float*)d_in[0];
    const float* pts = (const float*)d_in[1];
    float* out  = (float*)d_out;
    float* nxyz = (float*)d_ws;                       // (B, S, 3) sampled centers

    fps_kernel<<<BATCH, 1024, 0, stream>>>(xyz, nxyz, out);

    auto LP = [&](int br, int ly) -> LayerP {
        const int base = 2 + br * 18 + ly * 6;        // w,b,gamma,beta,mean,var
        return LayerP{(const float*)d_in[base],     (const float*)d_in[base + 1],
                      (const float*)d_in[base + 2], (const float*)d_in[base + 3],
                      (const float*)d_in[base + 4], (const float*)d_in[base + 5]};
    };
    BranchP P0{LP(0, 0), LP(0, 1), LP(0, 2)};
    BranchP P1{LP(1, 0), LP(1, 1), LP(1, 2)};
    BranchP P2{LP(2, 0), LP(2, 1), LP(2, 2)};

    msg_branch<16, 32, 32, 64>
        <<<BATCH * S_CENT / 8, 256, 0, stream>>>(xyz, pts, nxyz, out, P0, 0.01f, 0);
    msg_branch<32, 64, 64, 128>
        <<<BATCH * S_CENT / 4, 256, 0, stream>>>(xyz, pts, nxyz, out, P1, 0.04f, 64);
    msg_branch<128, 64, 96, 128>
        <<<BATCH * S_CENT, 256, 0, stream>>>(xyz, pts, nxyz, out, P2, 0.16f, 192);
}